// MultiHeadAttentionCross_40441412059498
// MI455X (gfx1250) — compile-verified
//
#include <hip/hip_runtime.h>
#include <math.h>

#define B_ 16
#define C_ 512
#define H_ 8
#define TT_ 1024
#define TS_ 1024
#define KC_ 64
#define ENH_SCORE_ 1.0f

typedef __attribute__((ext_vector_type(2))) float v2f;
typedef __attribute__((ext_vector_type(8))) float v8f;

// D = A(16x4, f32) * B(4x16, f32) + C(16x16, f32), exact fp32 WMMA
__device__ __forceinline__ v8f wmma_f32(v2f a, v2f b, v8f c) {
  return __builtin_amdgcn_wmma_f32_16x16x4_f32(false, a, false, b, (short)0, c,
                                               false, false);
}

// ---------------------------------------------------------------------------
// Kernel 1: O[b,o,t] = sum_c W[o,c] * X[b,c,t] + bias[o]; optional RoPE on
// first 32 dims of each 64-wide head; scatter to head layout R[b,h,t,d].
// Each wave owns two stacked 16-row o-tiles (o0..o0+15, o0+16..o0+31) so the
// RoPE pair (d, d+16) is element-aligned across the two accumulators.
// ---------------------------------------------------------------------------
__global__ __launch_bounds__(256) void proj_rope_kernel(
    const float* __restrict__ W, const float* __restrict__ bias,
    const float* __restrict__ X, float* __restrict__ R, int T, int do_rope) {
  const int lane = threadIdx.x & 31;
  const int wv   = threadIdx.x >> 5;
  const int t0   = blockIdx.x * 128 + wv * 16;
  const int o0   = blockIdx.y * 32;
  const int b    = blockIdx.z;
  const int row  = lane & 15;
  const int hi   = lane >> 4;   // 0: K=0..1, 1: K=2..3 (ISA A/B frag layout)
  const int idx2 = hi << 1;

  const float* Xb = X + (size_t)b * C_ * T;
  v8f c0 = {}, c1 = {};
  for (int kk = 0; kk < C_; kk += 4) {
    const int kA = kk + idx2;
    v2f a0, a1, bb;
    a0.x = W[(o0 + row) * C_ + kA];
    a0.y = W[(o0 + row) * C_ + kA + 1];
    a1.x = W[(o0 + 16 + row) * C_ + kA];
    a1.y = W[(o0 + 16 + row) * C_ + kA + 1];
    bb.x = Xb[(size_t)kA * T + t0 + row];          // coalesced over t
    bb.y = Xb[(size_t)(kA + 1) * T + t0 + row];
    c0 = wmma_f32(a0, bb, c0);
    c1 = wmma_f32(a1, bb, c1);
  }

  const int n = lane & 15;
  const int t = t0 + n;
  const bool rope_half = do_rope && ((o0 & 32) == 0);  // d in [0,32)
  for (int j = 0; j < 8; ++j) {
    const int m  = j + (hi << 3);
    const int oA = o0 + m;        // frag0: d = m        (when o0%64==0)
    const int oB = o0 + 16 + m;   // frag1: d = m + 16
    float v0 = c0[j] + bias[oA];
    float v1 = c1[j] + bias[oB];
    if (rope_half) {
      // 10000^(-m/16) == exp2(-m * log2(10000)/16); single TRANS op vs powf
      const float theta = exp2f((float)m * -0.83048202372184059f);
      float sn, cs;
      sincosf((float)t * theta, &sn, &cs);
      const float n0 = v0 * cs - v1 * sn;   // x[d]*cos - x[d+16]*sin
      const float n1 = v1 * cs + v0 * sn;   // x[d+16]*cos + x[d]*sin
      v0 = n0; v1 = n1;
    }
    R[(((size_t)b * H_ + (oA >> 6)) * T + t) * KC_ + (oA & 63)] = v0;
    R[(((size_t)b * H_ + (oB >> 6)) * T + t) * KC_ + (oB & 63)] = v1;
  }
}

// ---------------------------------------------------------------------------
// Kernel 2: fused attention for one (b, h, 32 q-rows) tile.
//   scores = (q k^T) * 1/8  -> softmax -> enhancement block := 1.0
//   -> write attn (coalesced) -> out = attn @ v, all with f32 WMMA.
// LDS: sS[32][1028] score/prob tile (pitch-padded: lanes hit distinct banks
// for column reads), sK[64][132] staged k^T chunk (reused as reduce scratch).
// Next k^T chunk is prefetched (global_prefetch_b8) while current computes.
// ---------------------------------------------------------------------------
__global__ __launch_bounds__(256) void attn_kernel(
    const float* __restrict__ qr, const float* __restrict__ kr,
    const float* __restrict__ vr, float* __restrict__ attn_out,
    float* __restrict__ outh,
    const int* __restrict__ p_syn_i, const int* __restrict__ p_syn_n,
    const int* __restrict__ p_ref_j, const int* __restrict__ p_ref_m) {
  extern __shared__ float sm[];
  const int SP = TS_ + 4;       // 1028
  const int KP = 132;
  float* sS = sm;               // 32 * 1028
  float* sK = sm + 32 * SP;     // 64 * 132

  const int lane = threadIdx.x & 31;
  const int wv   = threadIdx.x >> 5;
  const int tid  = threadIdx.x;
  const int h = blockIdx.y, b = blockIdx.z;
  const int Q0 = blockIdx.x * 32;
  const size_t bh = (size_t)b * H_ + h;

  const int row  = lane & 15;
  const int hi   = lane >> 4;
  const int idx2 = hi << 1;

  const int qs = wv & 1;        // which 16 q-rows of the 32
  const int pr = wv >> 1;       // 0..3
  const int q0 = qs * 16;

  // Preload q A-fragments for all 16 K-steps (KC=64): 32 VGPRs, reused for
  // every key tile of every chunk. Fold in softmax scale 1/sqrt(64)=0.125:
  // power-of-two => bit-exact vs scaling the accumulated scores.
  v2f aq[16];
  {
    const float* qrow = qr + (bh * TT_ + Q0 + q0 + row) * KC_;
    for (int k4 = 0; k4 < 16; ++k4) {
      aq[k4].x = qrow[k4 * 4 + idx2] * 0.125f;
      aq[k4].y = qrow[k4 * 4 + idx2 + 1] * 0.125f;
    }
  }

  // Phase 1: scores tile (32 x 1024) into LDS, chunked over 128 keys.
  const float* krb = kr + bh * TS_ * KC_;
  for (int kb = 0; kb < TS_ / 128; ++kb) {
    const int k0 = kb * 128;
    __syncthreads();  // previous chunk's readers done with sK
    for (int i = tid; i < 64 * 128; i += 256) {
      const int d = i & 63, key = i >> 6;     // coalesced read of k rows
      sK[d * KP + key] = krb[(size_t)(k0 + key) * KC_ + d];
    }
    // prefetch next chunk (128 key rows x 256B) while this chunk computes
    if (kb + 1 < TS_ / 128 && tid < 128) {
      __builtin_prefetch(krb + (size_t)(k0 + 128 + tid) * KC_, 0, 0);
    }
    __syncthreads();
    for (int sub = 0; sub < 2; ++sub) {
      const int kt = (pr * 2 + sub) * 16;     // key tile within chunk
      v8f acc = {};
      for (int k4 = 0; k4 < 16; ++k4) {
        v2f bb;
        bb.x = sK[(k4 * 4 + idx2) * KP + kt + row];
        bb.y = sK[(k4 * 4 + idx2 + 1) * KP + kt + row];
        acc = wmma_f32(aq[k4], bb, acc);
      }
      for (int j = 0; j < 8; ++j) {
        const int m = j + (hi << 3);
        sS[(q0 + m) * SP + k0 + kt + row] = acc[j];  // scale already in aq
      }
    }
  }
  __syncthreads();

  // Phase 2: row softmax (8 threads per row), then enhancement + attn store.
  float* sRed = sK;           // 256 partials
  float* sInv = sK + 256;     // 32 reciprocals
  float* sMax = sK + 288;     // 32 row maxima
  const int r  = tid >> 3;
  const int su = tid & 7;
  {
    float mx = -3.402823466e38f;
    const float* rp = sS + r * SP + su * 128;
    for (int i = 0; i < 128; ++i) mx = fmaxf(mx, rp[i]);
    sRed[r * 8 + su] = mx;
  }
  __syncthreads();
  if (su == 0) {
    float mx = sRed[r * 8];
    for (int i = 1; i < 8; ++i) mx = fmaxf(mx, sRed[r * 8 + i]);
    sMax[r] = mx;
  }
  __syncthreads();
  {
    const float mx = sMax[r];
    float* rp = sS + r * SP + su * 128;
    float s = 0.0f;
    for (int i = 0; i < 128; ++i) {
      const float e = expf(rp[i] - mx);
      rp[i] = e;
      s += e;
    }
    sRed[r * 8 + su] = s;
  }
  __syncthreads();
  if (su == 0) {
    float s = 0.0f;
    for (int i = 0; i < 8; ++i) s += sRed[r * 8 + i];
    sInv[r] = 1.0f / s;
  }
  __syncthreads();
  {
    const int syn_i = *p_syn_i, syn_n = *p_syn_n;
    const int ref_j = *p_ref_j, ref_m = *p_ref_m;
    float* ao = attn_out + (bh * TT_ + Q0) * TS_;
    for (int idx = tid; idx < 32 * TS_; idx += 256) {   // coalesced store
      const int rr = idx >> 10, col = idx & (TS_ - 1);
      float p = sS[rr * SP + col] * sInv[rr];
      const int qg = Q0 + rr;
      if (qg >= syn_i && qg < syn_i + syn_n && col >= ref_j &&
          col < ref_j + ref_m)
        p = p + (1.0f - p) * ENH_SCORE_;
      sS[rr * SP + col] = p;                 // attn@v uses enhanced values
      ao[(size_t)rr * TS_ + col] = p;
    }
  }
  __syncthreads();

  // Phase 3: out(32x64) = attn(32x1024) @ v(1024x64). A from LDS (pitch 1028
  // => distinct banks across lanes), B coalesced from global v (L2-resident).
  {
    const int d0 = pr * 16;
    const float* vb = vr + bh * TS_ * KC_;
    v8f acc = {};
    for (int kk = 0; kk < TS_; kk += 4) {
      v2f a, bb;
      a.x  = sS[(q0 + row) * SP + kk + idx2];
      a.y  = sS[(q0 + row) * SP + kk + idx2 + 1];
      bb.x = vb[(size_t)(kk + idx2) * KC_ + d0 + row];
      bb.y = vb[(size_t)(kk + idx2 + 1) * KC_ + d0 + row];
      acc = wmma_f32(a, bb, acc);
    }
    float* ob = outh + (bh * TT_ + Q0 + q0) * KC_ + d0;
    for (int j = 0; j < 8; ++j) {
      const int m = j + (hi << 3);
      ob[(size_t)m * KC_ + row] = acc[j];
    }
  }
}

// ---------------------------------------------------------------------------
// Kernel 3: y[b,o,t] = sum_c Wo[o,c] * outh[b, c>>6, t, c&63] + bo[o]
// ---------------------------------------------------------------------------
__global__ __launch_bounds__(256) void outproj_kernel(
    const float* __restrict__ Wo, const float* __restrict__ bo,
    const float* __restrict__ outh, float* __restrict__ Y) {
  const int lane = threadIdx.x & 31;
  const int wv   = threadIdx.x >> 5;
  const int t0   = blockIdx.x * 128 + wv * 16;
  const int o0   = blockIdx.y * 32;
  const int b    = blockIdx.z;
  const int row  = lane & 15;
  const int hi   = lane >> 4;
  const int idx2 = hi << 1;

  const float* ob = outh + (size_t)b * H_ * TT_ * KC_;
  v8f c0 = {}, c1 = {};
  for (int kk = 0; kk < C_; kk += 4) {
    const int kA = kk + idx2;
    v2f a0, a1, bb;
    a0.x = Wo[(o0 + row) * C_ + kA];
    a0.y = Wo[(o0 + row) * C_ + kA + 1];
    a1.x = Wo[(o0 + 16 + row) * C_ + kA];
    a1.y = Wo[(o0 + 16 + row) * C_ + kA + 1];
    bb.x = ob[((size_t)(kA >> 6) * TT_ + t0 + row) * KC_ + (kA & 63)];
    bb.y = ob[((size_t)((kA + 1) >> 6) * TT_ + t0 + row) * KC_ + ((kA + 1) & 63)];
    c0 = wmma_f32(a0, bb, c0);
    c1 = wmma_f32(a1, bb, c1);
  }
  const int n = lane & 15;
  const int t = t0 + n;
  for (int j = 0; j < 8; ++j) {
    const int m  = j + (hi << 3);
    const int oA = o0 + m, oB = o0 + 16 + m;
    Y[((size_t)b * C_ + oA) * TT_ + t] = c0[j] + bo[oA];
    Y[((size_t)b * C_ + oB) * TT_ + t] = c1[j] + bo[oB];
  }
}

extern "C" void kernel_launch(void* const* d_in, const int* in_sizes, int n_in,
                              void* d_out, int out_size, void* d_ws,
                              size_t ws_size, hipStream_t stream) {
  (void)in_sizes; (void)n_in; (void)out_size; (void)ws_size;
  const float* x  = (const float*)d_in[0];
  const float* c  = (const float*)d_in[1];
  const float* Wq = (const float*)d_in[2];
  const float* bq = (const float*)d_in[3];
  const float* Wk = (const float*)d_in[4];
  const float* bk = (const float*)d_in[5];
  const float* Wv = (const float*)d_in[6];
  const float* bv = (const float*)d_in[7];
  const float* Wo = (const float*)d_in[8];
  const float* bo = (const float*)d_in[9];
  const int* syn_i = (const int*)d_in[10];
  const int* syn_n = (const int*)d_in[11];
  const int* ref_j = (const int*)d_in[12];
  const int* ref_m = (const int*)d_in[13];

  const size_t QN = (size_t)B_ * H_ * TT_ * KC_;  // 8,388,608 floats
  float* qr = (float*)d_ws;
  float* kr = qr + QN;
  float* vr = kr + QN;
  float* oh = vr + QN;

  float* y    = (float*)d_out;
  float* attn = y + (size_t)B_ * C_ * TT_;

  dim3 blk(256);
  dim3 g1(TT_ / 128, C_ / 32, B_);
  proj_rope_kernel<<<g1, blk, 0, stream>>>(Wq, bq, x, qr, TT_, 1);
  proj_rope_kernel<<<g1, blk, 0, stream>>>(Wk, bk, c, kr, TS_, 1);
  proj_rope_kernel<<<g1, blk, 0, stream>>>(Wv, bv, c, vr, TS_, 0);

  const size_t smem =
      (size_t)(32 * (TS_ + 4) + 64 * 132) * sizeof(float);  // ~162 KB
  (void)hipFuncSetAttribute(reinterpret_cast<const void*>(attn_kernel),
                            hipFuncAttributeMaxDynamicSharedMemorySize,
                            (int)smem);
  dim3 g2(TT_ / 32, H_, B_);
  attn_kernel<<<g2, blk, smem, stream>>>(qr, kr, vr, attn, oh, syn_i, syn_n,
                                         ref_j, ref_m);

  outproj_kernel<<<g1, blk, 0, stream>>>(Wo, bo, oh, y);
}